// WRMDGC_50405736186369
// MI455X (gfx1250) — compile-verified
//
#include <hip/hip_runtime.h>
#include <hip/hip_bf16.h>
#include <math.h>

// Problem constants (from reference)
#define BB 32
#define TT 12
#define NN 512
#define CC 32
#define HH 64
#define DD 16
#define CH 96       // C + H
#define KI 192      // 2*(C+H)
#define OG 128      // 2*H
#define OC 64       // H
#define CHUNK 128   // K-chunk staged in LDS for A@X

typedef float v2f __attribute__((ext_vector_type(2)));
typedef float v8f __attribute__((ext_vector_type(8)));
typedef unsigned int v4u __attribute__((ext_vector_type(4)));
typedef int v4i __attribute__((ext_vector_type(4)));
typedef int v8i __attribute__((ext_vector_type(8)));

static __device__ __forceinline__ v8f wmma4(v2f a, v2f b, v8f c) {
  // V_WMMA_F32_16X16X4_F32 : D = A(16x4) * B(4x16) + C(16x16)
  return __builtin_amdgcn_wmma_f32_16x16x4_f32(
      /*neg_a=*/false, a, /*neg_b=*/false, b,
      /*c_mod=*/(short)0, c, /*reuse_a=*/false, /*reuse_b=*/false);
}

static __device__ __forceinline__ float sigm(float x) {
  return 1.0f / (1.0f + __expf(-x));
}

// ---------------------------------------------------------------------------
// 1) s_ref[r][bt][c] = mean_n |x - ref|
__global__ void k_absmean(const float* __restrict__ x,
                          const float* __restrict__ wk,
                          const float* __restrict__ dy,
                          const float* __restrict__ rc,
                          float* __restrict__ s3) {
  int bt = blockIdx.x;          // 0..B*T-1
  int r  = blockIdx.y;          // 0..2
  int c  = threadIdx.x;         // 0..31
  const float* ref = (r == 0) ? wk : (r == 1) ? dy : rc;
  size_t base = (size_t)bt * NN * CC + c;
  float acc = 0.f;
  for (int n = 0; n < NN; ++n) {
    size_t idx = base + (size_t)n * CC;
    acc += fabsf(x[idx] - ref[idx]);
  }
  s3[((size_t)r * (BB * TT) + bt) * CC + c] = acc * (1.0f / NN);
}

// 2) w[b,t,c] = sum_r s[r][b,t,c] / (sum_t' s[r][b,t',c] + 1e-6)
__global__ void k_weights(const float* __restrict__ s3, float* __restrict__ w) {
  int b = blockIdx.x;
  int t = threadIdx.y;          // 0..11
  int c = threadIdx.x;          // 0..31
  float acc = 0.f;
  for (int r = 0; r < 3; ++r) {
    float den = 0.f;
    for (int tp = 0; tp < TT; ++tp)
      den += s3[((size_t)r * (BB * TT) + b * TT + tp) * CC + c];
    acc += s3[((size_t)r * (BB * TT) + b * TT + t) * CC + c] / (den + 1e-6f);
  }
  w[((size_t)b * TT + t) * CC + c] = acc;
}

// 3) Wf[n][ki][o] = sum_d E[n][d] * Wp[d][k][i][o]   (ki = k*96+i)
__global__ void k_nodeW(const float* __restrict__ E, const float* __restrict__ Wp,
                        float* __restrict__ Wf, int O) {
  size_t idx = (size_t)blockIdx.x * blockDim.x + threadIdx.x;
  size_t total = (size_t)NN * KI * O;
  if (idx >= total) return;
  int o  = idx % O;
  int ki = (idx / O) % KI;
  int n  = idx / ((size_t)O * KI);
  int k = ki / CH, i = ki % CH;
  float acc = 0.f;
  for (int d = 0; d < DD; ++d)
    acc += E[n * DD + d] * Wp[(((size_t)d * 2 + k) * CH + i) * O + o];
  Wf[idx] = acc;
}

__global__ void k_nodeB(const float* __restrict__ E, const float* __restrict__ bp,
                        float* __restrict__ bf, int O) {
  size_t idx = (size_t)blockIdx.x * blockDim.x + threadIdx.x;
  if (idx >= (size_t)NN * O) return;
  int o = idx % O, n = idx / O;
  float acc = 0.f;
  for (int d = 0; d < DD; ++d) acc += E[n * DD + d] * bp[d * O + o];
  bf[idx] = acc;
}

// 4) state init, gmem zero
__global__ void k_init(const float* __restrict__ s0, float* __restrict__ st,
                       float* __restrict__ gm) {
  size_t idx = (size_t)blockIdx.x * blockDim.x + threadIdx.x;
  if (idx >= (size_t)BB * NN * HH) return;
  st[idx] = s0[idx];
  gm[idx] = 0.f;
}

// ---------------------------------------------------------------------------
// 5) A[b] = softmax(relu(Ed Ed^T)) : one wave computes a 16-row stripe via WMMA
__global__ void __launch_bounds__(32)
k_adj(const float* __restrict__ emb, int t, float* __restrict__ A) {
  int b = blockIdx.y, mt = blockIdx.x, lane = threadIdx.x;
  int hl = lane & 15, hi = lane >> 4;
  const float* Ed = emb + ((size_t)b * TT + t) * NN * DD;
  __shared__ float stripe[16][NN];       // 32 KB
  int m0 = mt * 16;
  // preload A fragments for K=16 (4 steps of 4)
  v2f afr[4];
  for (int q = 0; q < 4; ++q) {
    const float* p = Ed + (size_t)(m0 + hl) * DD + q * 4 + hi * 2;
    afr[q].x = p[0]; afr[q].y = p[1];
  }
  for (int ct = 0; ct < 32; ++ct) {
    v8f acc = {};
    int col = ct * 16 + hl;
    for (int q = 0; q < 4; ++q) {
      const float* p = Ed + (size_t)col * DD + q * 4 + hi * 2;
      v2f bf; bf.x = p[0]; bf.y = p[1];     // B[k][m] = Ed[m][k]
      acc = wmma4(afr[q], bf, acc);
    }
    for (int j = 0; j < 8; ++j) {
      float v = acc[j];
      stripe[j + hi * 8][ct * 16 + hl] = v > 0.f ? v : 0.f;   // relu
    }
  }
  __syncthreads();
  if (lane < 16) {
    float* rp = stripe[lane];
    float mx = -1e30f;
    for (int m = 0; m < NN; ++m) mx = fmaxf(mx, rp[m]);
    float sum = 0.f;
    for (int m = 0; m < NN; ++m) { float e = __expf(rp[m] - mx); rp[m] = e; sum += e; }
    float inv = 1.f / sum;
    float* Ar = A + ((size_t)b * NN + m0 + lane) * NN;
    for (int m = 0; m < NN; ++m) Ar[m] = rp[m] * inv;
  }
}

// 6) xcat[b][n][0:32] = cur_t ; [32:96] = state (mode 0) or z*state (mode 1)
__global__ void k_concat(const float* __restrict__ x, const float* __restrict__ w,
                         const float* __restrict__ state, const float* __restrict__ zr,
                         float* __restrict__ xcat, int t, int mode) {
  size_t idx = (size_t)blockIdx.x * blockDim.x + threadIdx.x;
  if (idx >= (size_t)BB * NN * CH) return;
  int i = idx % CH;
  size_t bn = idx / CH;
  int n = bn % NN, b = bn / NN;
  float v;
  if (i < CC) {
    v = x[(((size_t)b * TT + t) * NN + n) * CC + i] *
        (1.f + w[((size_t)b * TT + t) * CC + i]);
  } else {
    int o = i - CC;
    float s = state[((size_t)b * NN + n) * HH + o];
    if (mode == 1) s *= zr[((size_t)b * NN + n) * OG + o];   // z gate
    v = s;
  }
  xcat[idx] = v;
}

// 7) Y[b] = A[b] @ X[b]   (512x512 x 512x96).
//    4 waves/block, one 16-row stripe each; X panel staged in LDS chunks by the
//    Tensor Data Mover (double-buffered, overlapped with WMMA via TENSORcnt).
__global__ void __launch_bounds__(128)
k_gemm_Ax(const float* __restrict__ A, const float* __restrict__ X,
          float* __restrict__ Y) {
  int b    = blockIdx.y;
  int tid  = threadIdx.x;
  int wave = tid >> 5;
  int lane = tid & 31;
  int hl = lane & 15, hi = lane >> 4;
  int mt = blockIdx.x * 4 + wave;
  int m0 = mt * 16;
  const float* Ab = A + (size_t)b * NN * NN;
  const float* Xb = X + (size_t)b * NN * CH;
  float* Yb = Y + (size_t)b * NN * CH;

  __shared__ float Xs[2][CHUNK][CH];     // 2 x 48 KB = 96 KB

  // Issue a TDM load of chunk c (CHUNK x CH fp32 tile) into buffer bi.
  auto issue = [&](int c, int bi) {
    unsigned long long gaddr =
        (unsigned long long)(uintptr_t)(Xb + (size_t)c * CHUNK * CH);
    unsigned int ldsoff = (unsigned int)(uintptr_t)(&Xs[bi][0][0]);
    v4u g0;
    g0.x = 1u;                                   // count=1, user descriptor
    g0.y = ldsoff;                               // lds_addr (bytes)
    g0.z = (unsigned int)gaddr;                  // global_addr[31:0]
    g0.w = (unsigned int)((gaddr >> 32) & 0x01FFFFFFu) | 0x80000000u; // type=2
    v8i g1;
    g1[0] = 0x00020000;          // data_size=2 (4 bytes); no iterate/pad/mask
    g1[1] = (int)(CH << 16);     // tensor_dim0 = 96 (low half in [31:16])
    g1[2] = (int)(CHUNK << 16);  // tensor_dim1 = 128
    g1[3] = (int)(CH << 16);     // tile_dim0 = 96
    g1[4] = CHUNK;               // tile_dim1 = 128, tile_dim2 = 0
    g1[5] = CH;                  // tensor_dim0_stride = 96
    g1[6] = 0;
    g1[7] = 0;
    v4i z4 = {0, 0, 0, 0};
    v8i z8 = {0, 0, 0, 0, 0, 0, 0, 0};
    __builtin_amdgcn_tensor_load_to_lds(g0, g1, z4, z4, z8, 0);
  };

  if (wave == 0) {
    issue(0, 0);
    __builtin_amdgcn_s_wait_tensorcnt(0);
  }
  __syncthreads();

  v8f acc[6] = {};
  const float* arow = Ab + (size_t)(m0 + hl) * NN;
  for (int c = 0; c < NN / CHUNK; ++c) {
    int bi = c & 1;
    if (wave == 0 && c + 1 < NN / CHUNK) issue(c + 1, bi ^ 1);
    for (int kk = 0; kk < CHUNK; kk += 4) {
      v2f af;
      { const float* p = arow + c * CHUNK + kk + hi * 2; af.x = p[0]; af.y = p[1]; }
      int kr = kk + hi * 2;
#pragma unroll
      for (int nt = 0; nt < 6; ++nt) {
        int col = nt * 16 + hl;
        v2f bf;
        bf.x = Xs[bi][kr][col];
        bf.y = Xs[bi][kr + 1][col];
        acc[nt] = wmma4(af, bf, acc[nt]);
      }
    }
    __syncthreads();                         // all waves done reading Xs[bi]
    if (wave == 0 && c + 1 < NN / CHUNK)
      __builtin_amdgcn_s_wait_tensorcnt(0);  // next chunk landed
    __syncthreads();
  }
#pragma unroll
  for (int nt = 0; nt < 6; ++nt)
#pragma unroll
    for (int j = 0; j < 8; ++j)
      Yb[(size_t)(m0 + j + hi * 8) * CH + nt * 16 + hl] = acc[nt][j];
}

// 8) per-node GEMM: (32 x 192) @ Wn(192 x O) + bias, activation, scatter out.
//    Xin[b][ki] = ki<96 ? xcat[b][n][ki] : Ax[b][n][ki-96]
__global__ void __launch_bounds__(32)
k_nodegemm(const float* __restrict__ xcat, const float* __restrict__ Ax,
           const float* __restrict__ Wf, const float* __restrict__ bf,
           float* __restrict__ out, int O, int act) {
  int n = blockIdx.y, nt = blockIdx.x, lane = threadIdx.x;
  int hl = lane & 15, hi = lane >> 4;
  int ocol = nt * 16 + hl;
  const float* Wn = Wf + (size_t)n * KI * O;
  v8f acc0 = {}, acc1 = {};
  for (int kk = 0; kk < KI; kk += 4) {
    int ki = kk + hi * 2;
    v2f bfr;
    bfr.x = Wn[(size_t)ki * O + ocol];
    bfr.y = Wn[(size_t)(ki + 1) * O + ocol];
    const float* src = (ki < CH) ? xcat : Ax;
    int kio = (ki < CH) ? ki : (ki - CH);
    v2f a0, a1;
    { const float* p = src + ((size_t)hl * NN + n) * CH + kio;        a0.x = p[0]; a0.y = p[1]; }
    { const float* p = src + ((size_t)(16 + hl) * NN + n) * CH + kio; a1.x = p[0]; a1.y = p[1]; }
    acc0 = wmma4(a0, bfr, acc0);
    acc1 = wmma4(a1, bfr, acc1);
  }
  float bias = bf[(size_t)n * O + ocol];
#pragma unroll
  for (int j = 0; j < 8; ++j) {
    int b0 = j + hi * 8;
    int b1 = 16 + j + hi * 8;
    float v0 = acc0[j] + bias, v1 = acc1[j] + bias;
    if (act == 0) { v0 = sigm(v0); v1 = sigm(v1); }
    else          { v0 = tanhf(v0); v1 = tanhf(v1); }
    out[((size_t)b0 * NN + n) * O + ocol] = v0;
    out[((size_t)b1 * NN + n) * O + ocol] = v1;
  }
}

// 9) fused MLP time-gate + GRU state/gmem update + output write
__global__ void k_update(const float* __restrict__ time_, const float* __restrict__ day_,
                         const float* __restrict__ W1, const float* __restrict__ b1,
                         const float* __restrict__ W2, const float* __restrict__ b2,
                         const float* __restrict__ W3, const float* __restrict__ b3,
                         const float* __restrict__ zr, const float* __restrict__ cand,
                         float* __restrict__ state, float* __restrict__ gmem,
                         const float* __restrict__ ap, const float* __restrict__ bp,
                         float* __restrict__ out, int t) {
  size_t idx = (size_t)blockIdx.x * blockDim.x + threadIdx.x;
  if (idx >= (size_t)BB * NN) return;
  int n = idx % NN, b = idx / NN;
  float ti[2 * CC];
  size_t tb = (((size_t)b * TT + t) * NN + n) * CC;
  for (int c = 0; c < CC; ++c) { ti[c] = time_[tb + c]; ti[CC + c] = day_[tb + c]; }
  float h1[40];
  for (int j = 0; j < 40; ++j) {
    float s = b1[j];
    for (int c = 0; c < 2 * CC; ++c) s += ti[c] * W1[c * 40 + j];
    h1[j] = s > 0.f ? s : 0.f;
  }
  float h2[2];
  for (int k = 0; k < 2; ++k) {
    float s = b2[k];
    for (int j = 0; j < 40; ++j) s += h1[j] * W2[j * 2 + k];
    h2[k] = s > 0.f ? s : 0.f;
  }
  float tw = sigm(h2[0] * W3[0] + h2[1] * W3[1] + b3[0]);
  float av = ap[0], bv = bp[0];
  size_t sb = ((size_t)b * NN + n) * HH;
  size_t zb = ((size_t)b * NN + n) * OG;
  size_t ob = (((size_t)b * TT + t) * NN + n) * HH;
  for (int o = 0; o < HH; ++o) {
    float s   = state[sb + o];
    float r   = zr[zb + HH + o];
    float cd  = cand[sb + o];
    float s2  = r * s + (1.f - r) * cd;
    float g   = gmem[sb + o];
    float enh = tw * s2 + (1.f - tw) * g;
    float gn  = (1.f - av) * g + av * enh;
    float sn  = s2 + bv * enh;
    state[sb + o] = sn;
    gmem[sb + o]  = gn;
    out[ob + o]   = sn;
  }
}

// ---------------------------------------------------------------------------
extern "C" void kernel_launch(void* const* d_in, const int* in_sizes, int n_in,
                              void* d_out, int out_size, void* d_ws, size_t ws_size,
                              hipStream_t stream) {
  const float* x      = (const float*)d_in[0];
  const float* s0     = (const float*)d_in[1];
  const float* embD   = (const float*)d_in[2];
  const float* embS   = (const float*)d_in[3];
  const float* time_  = (const float*)d_in[4];
  const float* day_   = (const float*)d_in[5];
  const float* weekly = (const float*)d_in[8];
  const float* daily  = (const float*)d_in[9];
  const float* recent = (const float*)d_in[10];
  const float* Wg     = (const float*)d_in[11];
  const float* bg     = (const float*)d_in[12];
  const float* Wc     = (const float*)d_in[13];
  const float* bc     = (const float*)d_in[14];
  const float* W1     = (const float*)d_in[15];
  const float* b1     = (const float*)d_in[16];
  const float* W2     = (const float*)d_in[17];
  const float* b2     = (const float*)d_in[18];
  const float* W3     = (const float*)d_in[19];
  const float* b3     = (const float*)d_in[20];
  const float* ascl   = (const float*)d_in[21];
  const float* bscl   = (const float*)d_in[22];
  float* out = (float*)d_out;

  // workspace carve-up (floats)
  float* p = (float*)d_ws;
  float* s3    = p; p += 3 * BB * TT * CC;
  float* wbtc  = p; p += BB * TT * CC;
  float* Wgf   = p; p += (size_t)NN * KI * OG;
  float* Wcf   = p; p += (size_t)NN * KI * OC;
  float* bgf   = p; p += (size_t)NN * OG;
  float* bcf   = p; p += (size_t)NN * OC;
  float* Adj   = p; p += (size_t)BB * NN * NN;
  float* state = p; p += (size_t)BB * NN * HH;
  float* gmem  = p; p += (size_t)BB * NN * HH;
  float* xcat  = p; p += (size_t)BB * NN * CH;
  float* Axc   = p; p += (size_t)BB * NN * CH;
  float* zr    = p; p += (size_t)BB * NN * OG;
  float* cand  = p; p += (size_t)BB * NN * OC;

  // ---- time-invariant precompute ----
  k_absmean<<<dim3(BB * TT, 3), 32, 0, stream>>>(x, weekly, daily, recent, s3);
  k_weights<<<BB, dim3(CC, TT), 0, stream>>>(s3, wbtc);
  {
    size_t tg = (size_t)NN * KI * OG;
    k_nodeW<<<(tg + 255) / 256, 256, 0, stream>>>(embS, Wg, Wgf, OG);
    size_t tc = (size_t)NN * KI * OC;
    k_nodeW<<<(tc + 255) / 256, 256, 0, stream>>>(embS, Wc, Wcf, OC);
    k_nodeB<<<((size_t)NN * OG + 255) / 256, 256, 0, stream>>>(embS, bg, bgf, OG);
    k_nodeB<<<((size_t)NN * OC + 255) / 256, 256, 0, stream>>>(embS, bc, bcf, OC);
  }
  k_init<<<((size_t)BB * NN * HH + 255) / 256, 256, 0, stream>>>(s0, state, gmem);

  const size_t nCat = (size_t)BB * NN * CH;
  // ---- recurrent steps ----
  for (int t = 0; t < TT; ++t) {
    k_adj<<<dim3(NN / 16, BB), 32, 0, stream>>>(embD, t, Adj);
    // gate
    k_concat<<<(nCat + 255) / 256, 256, 0, stream>>>(x, wbtc, state, zr, xcat, t, 0);
    k_gemm_Ax<<<dim3(NN / 64, BB), 128, 0, stream>>>(Adj, xcat, Axc);
    k_nodegemm<<<dim3(OG / 16, NN), 32, 0, stream>>>(xcat, Axc, Wgf, bgf, zr, OG, 0);
    // candidate
    k_concat<<<(nCat + 255) / 256, 256, 0, stream>>>(x, wbtc, state, zr, xcat, t, 1);
    k_gemm_Ax<<<dim3(NN / 64, BB), 128, 0, stream>>>(Adj, xcat, Axc);
    k_nodegemm<<<dim3(OC / 16, NN), 32, 0, stream>>>(xcat, Axc, Wcf, bcf, cand, OC, 1);
    // update
    k_update<<<((size_t)BB * NN + 127) / 128, 128, 0, stream>>>(
        time_, day_, W1, b1, W2, b2, W3, b3, zr, cand, state, gmem, ascl, bscl, out, t);
  }
}